// Loss_8005819040200
// MI455X (gfx1250) — compile-verified
//
#include <hip/hip_runtime.h>
#include <hip/hip_bf16.h>

typedef __attribute__((ext_vector_type(2))) float v2f;
typedef __attribute__((ext_vector_type(8))) float v8f;
typedef __attribute__((ext_vector_type(4))) unsigned int u32x4;
typedef __attribute__((ext_vector_type(8))) int i32x8;
typedef __attribute__((ext_vector_type(4))) int i32x4;

#define BATCH 2
#define NUM_POINT 1024
#define UP 8
#define N_GEN (NUM_POINT * UP)
#define N_DENSE 8192
#define NQ 8192                    // both chamfer directions have 8192 columns

// ---------------- workspace layout (floats) ----------------
// acc[0] = chamfer sum   acc[1] = reg sum
// acc[2] = Sg            acc[3] = Suv
// acc[4] = arap sum      acc[5] = overlap sum
// ws+16           : uv_d [BATCH*NUM_POINT*32]
// ws+16+65536     : g_d  [BATCH*NUM_POINT*32]

__global__ void zero_accum(float* acc) {
    if (threadIdx.x < 8) acc[threadIdx.x] = 0.0f;
}

// ---------------------------------------------------------------------------
// Chamfer: one wave per 16-row block of P; min over all NQ columns of Q.
// dist(m,n) = |P_m|^2 + |Q_n|^2 - 2 P_m.Q_n.  The WMMA computes the fused
// term C = -2*dot + |Q_n|^2 by packing A'=(-2p,1), B'=(q,|q|^2) into K=4.
// Q (96KB) is staged once per workgroup into LDS by the Tensor Data Mover,
// overlapped with per-wave P-row setup.
// ---------------------------------------------------------------------------
__global__ void __launch_bounds__(256)
chamfer_rowmin(const float* __restrict__ P, int nP,
               const float* __restrict__ Q,
               float* __restrict__ accum) {
    __shared__ float ldsQ[NQ * 3];          // 96 KB, fits CDNA5 320KB WGP LDS

    const int lane = threadIdx.x & 31;
    const int wid  = threadIdx.x >> 5;
    const int rb   = blockIdx.x * (blockDim.x >> 5) + wid;   // 16-row block
    const int b    = blockIdx.y;

    const float* Pb = P + (size_t)b * nP * 3;
    const float* Qb = Q + (size_t)b * NQ * 3;

    const int m  = lane & 15;
    const int hi = lane >> 4;                 // 0 -> K=0,1 ; 1 -> K=2,3

    // ---- per-wave P-row setup (overlaps the TDM DMA below) ----
    const float* pm = Pb + (size_t)(rb * 16 + m) * 3;
    const float px = pm[0], py = pm[1], pz = pm[2];

    v2f a;                                    // A' = (-2px,-2py,-2pz, 1)
    a.x = hi ? (-2.0f * pz) : (-2.0f * px);
    a.y = hi ? 1.0f : (-2.0f * py);

    const float pn = px * px + py * py + pz * pz;
    float rowpn[8];
#pragma unroll
    for (int r = 0; r < 8; ++r)
        rowpn[r] = __shfl(pn, r + 8 * hi, 32);   // |P|^2 for the 8 C-rows

    // ---- stage all of Q into LDS via TDM (one wave issues the DMA) ----
    if (wid == 0) {
        const unsigned long long ga = (unsigned long long)(uintptr_t)Qb;
        u32x4 g0;
        g0.x = 1u;                                    // count=1, no gather
        g0.y = 0u;                                    // lds_addr = 0
        g0.z = (unsigned int)ga;                      // global_addr[31:0]
        g0.w = (unsigned int)((ga >> 32) & 0x1FFFFFFu) | (2u << 30); // type=2
        i32x8 g1;
        g1[0] = (int)(2u << 16);                      // data_size = 4B
        g1[1] = (int)0x60000000u;                     // tensor_dim0 = 24576
        g1[2] = (int)0x00010000u;                     // tensor_dim1 = 1
        g1[3] = (int)0x60000000u;                     // tile_dim0   = 24576
        g1[4] = 1;                                    // tile_dim1   = 1
        g1[5] = 24576;                                // tensor_dim0_stride
        g1[6] = 0; g1[7] = 0;
        i32x4 gz4 = {0, 0, 0, 0};
        i32x8 gz8 = {0, 0, 0, 0, 0, 0, 0, 0};
        __builtin_amdgcn_tensor_load_to_lds(g0, g1, gz4, gz4, gz8, 0);
        __builtin_amdgcn_s_wait_tensorcnt(0);
    }
    __syncthreads();
    // The TDM builtin writes LDS behind the compiler's alias analysis; make
    // ldsQ escape so the loads below are not folded away.
    asm volatile("" : : "r"(&ldsQ[0]) : "memory");

    float rowmin[8];
#pragma unroll
    for (int r = 0; r < 8; ++r) rowmin[r] = 3.0e38f;

    int idx = m * 3;
#pragma unroll 4
    for (int cb = 0; cb < (NQ >> 4); ++cb, idx += 48) {
        const float qx = ldsQ[idx + 0], qy = ldsQ[idx + 1], qz = ldsQ[idx + 2];
        const float qn = qx * qx + qy * qy + qz * qz;

        v2f bv;                               // B' = (qx,qy,qz,|q|^2)
        bv.x = hi ? qz : qx;
        bv.y = hi ? qn : qy;

        v8f c = {};
        c = __builtin_amdgcn_wmma_f32_16x16x4_f32(false, a, false, bv,
                                                  (short)0, c, false, false);
#pragma unroll
        for (int r = 0; r < 8; ++r)
            rowmin[r] = fminf(rowmin[r], rowpn[r] + c[r]);
    }

    // min across the 16 lanes of each half (columns), stays within half
#pragma unroll
    for (int off = 1; off < 16; off <<= 1) {
#pragma unroll
        for (int r = 0; r < 8; ++r)
            rowmin[r] = fminf(rowmin[r], __shfl_xor(rowmin[r], off, 32));
    }
    if ((lane & 15) == 0) {                   // lane 0: rows 0..7, lane 16: 8..15
        float s = 0.0f;
#pragma unroll
        for (int r = 0; r < 8; ++r) s += rowmin[r];
        atomicAdd(accum + 0, s);
    }
}

// ---------------------------------------------------------------------------
// loss_on_reg
// ---------------------------------------------------------------------------
__global__ void reg_loss(const float* __restrict__ gen,
                         const float* __restrict__ sparse,
                         float* __restrict__ accum) {
    const int i = blockIdx.x * blockDim.x + threadIdx.x;   // 0 .. BATCH*N_GEN
    const int b = i / N_GEN, t = i % N_GEN;
    const float* g = gen + ((size_t)b * N_GEN + t) * 3;
    const float* s = sparse + ((size_t)b * NUM_POINT + t / UP) * 3;
    const float dx = s[0] - g[0], dy = s[1] - g[1], dz = s[2] - g[2];
    const float sq = dx * dx + dy * dy + dz * dz;
    const float sg = (sq > 0.04f) ? 1.5f : (sq < 0.04f ? 0.5f : 1.0f);
    float v = sq * sg;
#pragma unroll
    for (int off = 16; off >= 1; off >>= 1) v += __shfl_xor(v, off, 32);
    if ((threadIdx.x & 31) == 0) atomicAdd(accum + 1, v);
}

// ---------------------------------------------------------------------------
// loss_on_arap pass 1: per group (8 pts) 4-NN in UV, emit uv_d/g_d + sums
// ---------------------------------------------------------------------------
__global__ void arap_pass1(const float* __restrict__ gen,
                           const float* __restrict__ uv,
                           float* __restrict__ uvd_out,
                           float* __restrict__ gd_out,
                           float* __restrict__ accum) {
    const int gidx = blockIdx.x * blockDim.x + threadIdx.x;  // BATCH*NUM_POINT
    const float* g = gen + (size_t)gidx * UP * 3;
    const float* u = uv + (size_t)gidx * UP * 2;

    float ux[UP], uy[UP], gx[UP], gy[UP], gz[UP];
#pragma unroll
    for (int i = 0; i < UP; ++i) {
        ux[i] = u[i * 2 + 0]; uy[i] = u[i * 2 + 1];
        gx[i] = g[i * 3 + 0]; gy[i] = g[i * 3 + 1]; gz[i] = g[i * 3 + 2];
    }

    float sum_ud = 0.0f, sum_gd = 0.0f;
    for (int i = 0; i < UP; ++i) {
        float d[UP];
#pragma unroll
        for (int j = 0; j < UP; ++j) {
            const float du = ux[i] - ux[j], dv = uy[i] - uy[j];
            d[j] = du * du + dv * dv;
        }
        d[i] = 3.0e38f;                        // exclude self
        bool used[UP];
#pragma unroll
        for (int j = 0; j < UP; ++j) used[j] = false;

        for (int k = 0; k < 4; ++k) {
            float bd = 3.0e38f; int bj = 0;
#pragma unroll
            for (int j = 0; j < UP; ++j)
                if (!used[j] && d[j] < bd) { bd = d[j]; bj = j; }
            used[bj] = true;
            const float ud = sqrtf(bd + 1e-8f);
            const float ddx = gx[i] - gx[bj], ddy = gy[i] - gy[bj],
                        ddz = gz[i] - gz[bj];
            const float gd = sqrtf(ddx * ddx + ddy * ddy + ddz * ddz + 1e-8f);
            const int o = gidx * 32 + i * 4 + k;
            uvd_out[o] = ud; gd_out[o] = gd;
            sum_ud += ud; sum_gd += gd;
        }
    }
#pragma unroll
    for (int off = 16; off >= 1; off >>= 1) {
        sum_gd += __shfl_xor(sum_gd, off, 32);
        sum_ud += __shfl_xor(sum_ud, off, 32);
    }
    if ((threadIdx.x & 31) == 0) {
        atomicAdd(accum + 2, sum_gd);
        atomicAdd(accum + 3, sum_ud);
    }
}

// ---------------------------------------------------------------------------
// loss_on_arap pass 2: scale known, reduce sum of delta^2
// ---------------------------------------------------------------------------
__global__ void arap_pass2(const float* __restrict__ uvd,
                           const float* __restrict__ gd,
                           float* __restrict__ accum) {
    const int i = blockIdx.x * blockDim.x + threadIdx.x;  // BATCH*NUM_POINT*32
    const float scale = accum[2] / accum[3];
    const float delta = uvd[i] * scale - gd[i];
    float v = delta * delta;
#pragma unroll
    for (int off = 16; off >= 1; off >>= 1) v += __shfl_xor(v, off, 32);
    if ((threadIdx.x & 31) == 0) atomicAdd(accum + 4, v);
}

// ---------------------------------------------------------------------------
// loss_on_overlap: per sparse point, 6-NN among 1024, cross-min vs 8 gen pts
// ---------------------------------------------------------------------------
__global__ void overlap_loss(const float* __restrict__ sparse,
                             const float* __restrict__ gen,
                             float* __restrict__ accum) {
    const int i = blockIdx.x * blockDim.x + threadIdx.x;  // BATCH*NUM_POINT
    const int b = i / NUM_POINT, p = i % NUM_POINT;
    const float* S = sparse + (size_t)b * NUM_POINT * 3;
    const float sx = S[p * 3 + 0], sy = S[p * 3 + 1], sz = S[p * 3 + 2];

    float bestd[6]; int bestj[6];
#pragma unroll
    for (int k = 0; k < 6; ++k) { bestd[k] = 3.0e38f; bestj[k] = 0; }

    for (int j = 0; j < NUM_POINT; ++j) {
        if (j == p) continue;
        const float dx = sx - S[j * 3 + 0], dy = sy - S[j * 3 + 1],
                    dz = sz - S[j * 3 + 2];
        const float d = dx * dx + dy * dy + dz * dz;
        if (d < bestd[5]) {
            int k = 5;
            while (k > 0 && d < bestd[k - 1]) {
                bestd[k] = bestd[k - 1]; bestj[k] = bestj[k - 1]; --k;
            }
            bestd[k] = d; bestj[k] = j;
        }
    }

    const float* G = gen + ((size_t)b * NUM_POINT + p) * UP * 3;
    float s = 0.0f;
#pragma unroll
    for (int k = 0; k < 6; ++k) {
        const float* nb = S + (size_t)bestj[k] * 3;
        const float nx = nb[0], ny = nb[1], nz = nb[2];
        float mn = 3.0e38f;
#pragma unroll
        for (int uu = 0; uu < UP; ++uu) {
            const float dx = nx - G[uu * 3 + 0], dy = ny - G[uu * 3 + 1],
                        dz = nz - G[uu * 3 + 2];
            mn = fminf(mn, dx * dx + dy * dy + dz * dz);
        }
        s += mn;
    }
    atomicAdd(accum + 5, s);
}

// ---------------------------------------------------------------------------
__global__ void finalize(const float* __restrict__ acc, float* __restrict__ out) {
    if (threadIdx.x == 0 && blockIdx.x == 0) {
        const float total =
            (0.5f * acc[0] + acc[1] + acc[4] + acc[5]) / (float)BATCH;
        out[0] = total;
    }
}

// ---------------------------------------------------------------------------
extern "C" void kernel_launch(void* const* d_in, const int* in_sizes, int n_in,
                              void* d_out, int out_size, void* d_ws, size_t ws_size,
                              hipStream_t stream) {
    const float* gen    = (const float*)d_in[0];  // [B, N_GEN, 3]
    const float* uv     = (const float*)d_in[2];  // [B, N_GEN, 2]
    const float* sparse = (const float*)d_in[3];  // [B, NUM_POINT, 3]
    const float* dense  = (const float*)d_in[5];  // [B, N_DENSE, 3]
    float* out = (float*)d_out;

    float* acc = (float*)d_ws;
    float* uvd = acc + 16;
    float* gd  = uvd + (size_t)BATCH * NUM_POINT * 32;

    zero_accum<<<1, 32, 0, stream>>>(acc);

    // Chamfer, both directions (8 waves of 16-row blocks per 256-thread block)
    chamfer_rowmin<<<dim3(N_GEN / 16 / 8, BATCH), 256, 0, stream>>>(
        gen, N_GEN, dense, acc);
    chamfer_rowmin<<<dim3(N_DENSE / 16 / 8, BATCH), 256, 0, stream>>>(
        dense, N_DENSE, gen, acc);

    reg_loss<<<(BATCH * N_GEN) / 256, 256, 0, stream>>>(gen, sparse, acc);

    arap_pass1<<<(BATCH * NUM_POINT) / 128, 128, 0, stream>>>(gen, uv, uvd, gd, acc);
    arap_pass2<<<(BATCH * NUM_POINT * 32) / 256, 256, 0, stream>>>(uvd, gd, acc);

    overlap_loss<<<(BATCH * NUM_POINT) / 128, 128, 0, stream>>>(sparse, gen, acc);

    finalize<<<1, 64, 0, stream>>>(acc, out);
}